// TransformedReLU_46454366273944
// MI455X (gfx1250) — compile-verified
//
#include <hip/hip_runtime.h>

// ---------------------------------------------------------------------------
// Zonotope ReLU transform for MI455X (gfx1250, wave32).
//   x: [2048, 4096] f32  ->  out: [6144, 4096] f32
// Memory-bound: ~128 MiB HBM traffic -> ~5.5us roofline at 23.3 TB/s.
// Column |.| reduction is done on the WMMA pipe (V_WMMA_F32_16X16X4_F32,
// A = ones so D rows replicate the column sums of B = |x| tiles).
// ---------------------------------------------------------------------------

typedef __attribute__((ext_vector_type(2))) float v2f;
typedef __attribute__((ext_vector_type(8))) float v8f;

constexpr int kTerms        = 2048;              // N_TERMS
constexpr int kN            = 4096;              // neurons
constexpr int kSplits       = 16;                // row splits for reduction
constexpr int kRowsPerSplit = kTerms / kSplits;  // 128 rows (incl. row 0; fixed later)
constexpr int kColsPerBlock = 128;               // 8 waves * 16 cols

// ---- Kernel 1: partial column sums of |x| via WMMA ------------------------
// Wave handles 16 columns. B tile is 4x16 (K x N): VGPR0 = {lanes0-15:K=0,
// lanes16-31:K=2}, VGPR1 = {K=1, K=3}. A = ones(16x4) -> D[m][n] = col sum,
// replicated over m. Lane n<16 reads D VGPR0 for column c0+n.
__global__ __launch_bounds__(256)
void abs_colsum_wmma(const float* __restrict__ x, float* __restrict__ partials) {
  const int lane = threadIdx.x & 31;
  const int wave = threadIdx.x >> 5;
  const int col  = blockIdx.x * kColsPerBlock + wave * 16 + (lane & 15);
  // lanes 0-15 feed K rows {0,1}, lanes 16-31 feed K rows {2,3}
  const int r0   = blockIdx.y * kRowsPerSplit + ((lane >> 4) << 1);
  const float* p = x + (size_t)r0 * kN + col;

  v2f a; a[0] = 1.0f; a[1] = 1.0f;   // ones matrix (uniform across lanes)
  v8f acc = {};
  #pragma unroll 4
  for (int r = 0; r < kRowsPerSplit; r += 4) {
    v2f b;
    b[0] = __builtin_fabsf(p[(size_t)r * kN]);        // K = 0 / 2
    b[1] = __builtin_fabsf(p[(size_t)(r + 1) * kN]);  // K = 1 / 3
    acc = __builtin_amdgcn_wmma_f32_16x16x4_f32(
        /*neg_a=*/false, a, /*neg_b=*/false, b,
        /*c_mod=*/(short)0, acc, /*reuse_a=*/false, /*reuse_b=*/false);
  }
  if (lane < 16) partials[blockIdx.y * kN + col] = acc[0];
}

// ---- Kernel 2: per-column params + block-wide exclusive scan --------------
// Single block, 1024 threads, 4 columns each (in column order for the scan).
__global__ __launch_bounds__(1024)
void colparams(const float* __restrict__ x, const float* __restrict__ partials,
               float* __restrict__ out, float* __restrict__ scale,
               float* __restrict__ dhalf, int* __restrict__ trow) {
  __shared__ int sdata[1024];
  const int t = threadIdx.x;

  float scv[4], dhv[4], cenv[4];
  int   crv[4];
  int   cnt = 0;
  #pragma unroll
  for (int j = 0; j < 4; ++j) {
    const int c = t * 4 + j;
    float s = 0.0f;
    #pragma unroll
    for (int k = 0; k < kSplits; ++k) s += partials[k * kN + c];
    const float a0    = x[c];                       // row 0 (center)
    const float absum = s - __builtin_fabsf(a0);    // exclude row 0 from sum
    const float up = a0 + absum;
    const float lo = a0 - absum;
    const bool cross = (lo * up) < 0.0f;
    const bool pos   = lo >= 0.0f;                  // mutually exclusive w/ cross
    const float lam  = pos ? 1.0f : (cross ? up / (up - lo) : 0.0f);
    const float dh   = cross ? 0.5f * fmaxf(-lam * lo, (1.0f - lam) * up) : 0.0f;
    scv[j]  = cross ? lam : (pos ? 1.0f : 0.0f);
    cenv[j] = cross ? (dh + lam * a0) : (pos ? a0 : 0.0f);
    dhv[j]  = dh;
    crv[j]  = cross ? 1 : 0;
    cnt    += crv[j];
  }

  // Hillis-Steele inclusive scan over per-thread crossing counts.
  sdata[t] = cnt;
  __syncthreads();
  for (int off = 1; off < 1024; off <<= 1) {
    const int v = (t >= off) ? sdata[t - off] : 0;
    __syncthreads();
    sdata[t] += v;
    __syncthreads();
  }
  int run = kTerms + sdata[t] - cnt;  // exclusive prefix -> first new row id

  #pragma unroll
  for (int j = 0; j < 4; ++j) {
    const int c = t * 4 + j;
    out[c]   = cenv[j];               // output row 0 written here
    scale[c] = scv[j];
    dhalf[c] = dhv[j];
    trow[c]  = crv[j] ? run : -1;
    run     += crv[j];
  }
}

// ---- Kernel 3: rows 1..2047 = x * scale[col], float4 streaming ------------
__global__ __launch_bounds__(256)
void gens(const float* __restrict__ x, const float* __restrict__ scale,
          float* __restrict__ out) {
  const size_t i4   = (size_t)blockIdx.x * blockDim.x + threadIdx.x;
  const size_t base = (size_t)kN + i4 * 4;          // skip row 0
  const int    c    = (int)(base & (kN - 1));       // col of first elem (mult of 4)
  const float4 v = *reinterpret_cast<const float4*>(x + base);
  const float4 s = *reinterpret_cast<const float4*>(scale + c);
  float4 o;
  o.x = v.x * s.x; o.y = v.y * s.y; o.z = v.z * s.z; o.w = v.w * s.w;
  *reinterpret_cast<float4*>(out + base) = o;
}

// ---- Kernel 4: rows 2048..6143 = scatter(delta/2) else 0, float4 ----------
__global__ __launch_bounds__(256)
void tail(const float* __restrict__ dhalf, const int* __restrict__ trow,
          float* __restrict__ out) {
  const size_t i4   = (size_t)blockIdx.x * blockDim.x + threadIdx.x;
  const size_t elem = i4 * 4;
  const int r = kTerms + (int)(elem >> 12);         // elem / 4096
  const int c = (int)(elem & (kN - 1));
  const int4   tr = *reinterpret_cast<const int4*>(trow + c);
  const float4 dh = *reinterpret_cast<const float4*>(dhalf + c);
  float4 o;
  o.x = (tr.x == r) ? dh.x : 0.0f;
  o.y = (tr.y == r) ? dh.y : 0.0f;
  o.z = (tr.z == r) ? dh.z : 0.0f;
  o.w = (tr.w == r) ? dh.w : 0.0f;
  *reinterpret_cast<float4*>(out + (size_t)kTerms * kN + elem) = o;
}

extern "C" void kernel_launch(void* const* d_in, const int* in_sizes, int n_in,
                              void* d_out, int out_size, void* d_ws, size_t ws_size,
                              hipStream_t stream) {
  (void)in_sizes; (void)n_in; (void)out_size; (void)ws_size;
  const float* x   = (const float*)d_in[0];
  float*       out = (float*)d_out;

  // Workspace layout (all rewritten every call; no init needed):
  //   partials[16*4096] | scale[4096] | dhalf[4096] | trow[4096] (~310 KB)
  float* partials = (float*)d_ws;
  float* scale    = partials + (size_t)kSplits * kN;
  float* dhalf    = scale + kN;
  int*   trow     = (int*)(dhalf + kN);

  dim3 g1(kN / kColsPerBlock, kSplits);             // 32 x 16 blocks
  abs_colsum_wmma<<<g1, 256, 0, stream>>>(x, partials);

  colparams<<<1, 1024, 0, stream>>>(x, partials, out, scale, dhalf, trow);

  gens<<<((kTerms - 1) * kN / 4) / 256, 256, 0, stream>>>(x, scale, out);

  tail<<<((size_t)kN * kN / 4) / 256, 256, 0, stream>>>(dhalf, trow, out);
}